// SSDLoss_82411832476091
// MI455X (gfx1250) — compile-verified
//
#include <hip/hip_runtime.h>
#include <hip/hip_bf16.h>

#define B_  64
#define N_  8732
#define M_  32
#define C_  21
#define NTILE 35           // ceil(8732/256)
#define NPART (B_ * NTILE) // 2240 block partials from match kernel
#define NRED  (NPART + B_) // + 64 per-batch conf_neg = 2304 = 36*64

typedef float v2f __attribute__((ext_vector_type(2)));
typedef float v8f __attribute__((ext_vector_type(8)));

typedef __attribute__((address_space(1))) int g_i32;   // global (device) int
typedef __attribute__((address_space(3))) int l_i32;   // LDS int

// Detect CDNA5 async global->LDS builtins (toolchain-dependent availability)
#if defined(__has_builtin)
#  if __has_builtin(__builtin_amdgcn_global_load_async_to_lds_b32)
#    define HAVE_ASYNC_LDS 1
#  endif
#  if __has_builtin(__builtin_amdgcn_s_wait_asynccnt)
#    define HAVE_WAIT_ASYNC 1
#  endif
#endif

// ---------------------------------------------------------------- utilities
__device__ __forceinline__ unsigned f2key(float v) {
    unsigned u = __float_as_uint(v);
    return (u & 0x80000000u) ? ~u : (u | 0x80000000u);   // monotonic total order
}
__device__ __forceinline__ float key2f(unsigned k) {
    return __uint_as_float((k & 0x80000000u) ? (k ^ 0x80000000u) : ~k);
}

// ---------------------------------------------------------------- kernel 0
__global__ void zero_numpos(int* num_pos) {
    if (threadIdx.x < B_) num_pos[threadIdx.x] = 0;
}

// ---------------------------------------------------------------- kernel 1
// Fused: IoU match (GT boxes cached in LDS) + smooth-L1 loc loss +
// log-softmax over 21 classes + positive CE + background CE / ranking value.
__global__ void ssd_match_kernel(const float* __restrict__ pred_loc,
                                 const float* __restrict__ pred_cls,
                                 const float* __restrict__ gt_boxes,
                                 const int*   __restrict__ gt_labels,
                                 const float* __restrict__ dboxes,
                                 float* __restrict__ ranked,    // B*N
                                 int*   __restrict__ num_pos,   // B
                                 float* __restrict__ partials)  // NPART
{
    const int b    = blockIdx.y;
    const int tile = blockIdx.x;
    const int tid  = threadIdx.x;

    __shared__ float4 s_gt[M_];
    __shared__ int    s_lab[M_];
    __shared__ float  s_red[256];
    __shared__ int    s_pc;

    if (tid < M_) {
        s_gt[tid]  = ((const float4*)gt_boxes)[b * M_ + tid];
        s_lab[tid] = gt_labels[b * M_ + tid];
    }
    if (tid == 0) s_pc = 0;
    __syncthreads();

    const int n = tile * 256 + tid;
    float contrib = 0.0f;
    int   ispos   = 0;

    if (n < N_) {
        float4 d = ((const float4*)dboxes)[n];
        const float dl = d.x - 0.5f * d.z, dt = d.y - 0.5f * d.w;
        const float dr = d.x + 0.5f * d.z, db = d.y + 0.5f * d.w;
        const float area_d = d.z * d.w;

        float best = -1.0f; int bi = 0;
        #pragma unroll 8
        for (int m = 0; m < M_; ++m) {
            float4 g = s_gt[m];
            float gl = g.x - 0.5f * g.z, gt = g.y - 0.5f * g.w;
            float gr = g.x + 0.5f * g.z, gb = g.y + 0.5f * g.w;
            float iw = fmaxf(fminf(dr, gr) - fmaxf(dl, gl), 0.0f);
            float ih = fmaxf(fminf(db, gb) - fmaxf(dt, gt), 0.0f);
            float inter = iw * ih;
            float iou = inter / (area_d + g.z * g.w - inter + 1e-9f);
            if (iou > best) { best = iou; bi = m; }   // first max, like argmax
        }
        ispos = (best > 0.5f) ? 1 : 0;

        // ---- smooth-L1 localization loss vs encoded match
        float4 g = s_gt[bi];
        float gh0 = (g.x - d.x) / d.z;
        float gh1 = (g.y - d.y) / d.w;
        float gh2 = __logf(g.z / d.z);
        float gh3 = __logf(g.w / d.w);
        float4 p = ((const float4*)pred_loc)[(size_t)b * N_ + n];
        float lsum = 0.0f;
        {
            float df0 = p.x - gh0, df1 = p.y - gh1, df2 = p.z - gh2, df3 = p.w - gh3;
            float a0 = fabsf(df0), a1 = fabsf(df1), a2 = fabsf(df2), a3 = fabsf(df3);
            lsum += (a0 < 1.0f) ? 0.5f * df0 * df0 : a0 - 0.5f;
            lsum += (a1 < 1.0f) ? 0.5f * df1 * df1 : a1 - 0.5f;
            lsum += (a2 < 1.0f) ? 0.5f * df2 * df2 : a2 - 0.5f;
            lsum += (a3 < 1.0f) ? 0.5f * df3 * df3 : a3 - 0.5f;
        }

        // ---- log-softmax over 21 classes
        const float* cls = pred_cls + ((size_t)b * N_ + n) * C_;
        if (n + 256 < N_) __builtin_prefetch(cls + 256 * C_, 0, 0); // global_prefetch_b8
        float x[C_];
        float mx = -1e30f;
        #pragma unroll
        for (int c = 0; c < C_; ++c) { x[c] = cls[c]; mx = fmaxf(mx, x[c]); }
        float se = 0.0f;
        #pragma unroll
        for (int c = 0; c < C_; ++c) se += __expf(x[c] - mx);
        float lse = mx + __logf(se);

        float bg_ce  = lse - x[0];
        float pos_ce = lse - x[s_lab[bi]];

        contrib = ispos ? (lsum + pos_ce) : 0.0f;
        ranked[(size_t)b * N_ + n] = ispos ? -1.0e9f : bg_ce;
    }

    // ---- deterministic block reductions
    s_red[tid] = contrib;
    if (ispos) atomicAdd(&s_pc, 1);        // int atomics: deterministic
    __syncthreads();
    #pragma unroll
    for (int s = 128; s > 0; s >>= 1) {
        if (tid < s) s_red[tid] += s_red[tid + s];
        __syncthreads();
    }
    if (tid == 0) {
        partials[b * NTILE + tile] = s_red[0];
        if (s_pc) atomicAdd(&num_pos[b], s_pc);
    }
}

// ---------------------------------------------------------------- kernel 2
// Per-batch sum of top-k ranking values (k = min(3*np, N-np)) via 32-step
// binary search over order-preserving uint keys staged in LDS. Exact with ties.
// Staging uses CDNA5 GLOBAL_LOAD_ASYNC_TO_LDS (memory->LDS, no VGPR roundtrip)
// when the builtin is available, tracked with ASYNCcnt.
__global__ void hard_neg_topk_kernel(const float* __restrict__ ranked,
                                     const int*   __restrict__ num_pos,
                                     float* __restrict__ conf_neg) // B entries
{
    const int b   = blockIdx.x;
    const int tid = threadIdx.x;
    const int NT  = 1024;

    __shared__ unsigned s_key[N_];   // ~35 KB, fine in CDNA5's 320 KB/WGP LDS
    __shared__ float    s_red[NT];
    __shared__ int      s_cnt;

    const int np = num_pos[b];
    const int k  = min(3 * np, N_ - np);

#if defined(HAVE_ASYNC_LDS)
    // async DMA of the 8732 ranking floats straight into LDS
    for (int i = tid; i < N_; i += NT) {
        __builtin_amdgcn_global_load_async_to_lds_b32(
            (g_i32*)(ranked + (size_t)b * N_ + i),
            (l_i32*)(&s_key[i]),
            0, 0);
    }
#  if defined(HAVE_WAIT_ASYNC)
    __builtin_amdgcn_s_wait_asynccnt(0);
#  else
    asm volatile("s_wait_asynccnt 0x0" ::: "memory");
#  endif
    __syncthreads();
    // in-place float -> order-preserving key conversion
    for (int i = tid; i < N_; i += NT) {
        float v = __uint_as_float(s_key[i]);
        s_key[i] = f2key(v);
    }
    __syncthreads();
#else
    for (int i = tid; i < N_; i += NT)
        s_key[i] = f2key(ranked[(size_t)b * N_ + i]);
    __syncthreads();
#endif

    if (k <= 0) { if (tid == 0) conf_neg[b] = 0.0f; return; }

    // find T = key of the k-th largest element: max t with count(key >= t) >= k
    unsigned lo = 0u, hi = 0xFFFFFFFFu;
    while (lo < hi) {
        unsigned mid = (unsigned)((((unsigned long long)lo +
                                    (unsigned long long)hi) + 1ull) >> 1);
        if (tid == 0) s_cnt = 0;
        __syncthreads();
        int local = 0;
        for (int i = tid; i < N_; i += NT) local += (s_key[i] >= mid) ? 1 : 0;
        if (local) atomicAdd(&s_cnt, local);
        __syncthreads();
        const int cnt = s_cnt;
        if (cnt >= k) lo = mid; else hi = mid - 1;
        __syncthreads();
    }
    const unsigned Tk = lo;

    // sum of strictly-greater values + exact tie completion
    if (tid == 0) s_cnt = 0;
    __syncthreads();
    float lsum = 0.0f; int lcnt = 0;
    for (int i = tid; i < N_; i += NT) {
        unsigned u = s_key[i];
        if (u > Tk) { lsum += key2f(u); ++lcnt; }
    }
    s_red[tid] = lsum;
    if (lcnt) atomicAdd(&s_cnt, lcnt);
    __syncthreads();
    #pragma unroll
    for (int s = NT / 2; s > 0; s >>= 1) {
        if (tid < s) s_red[tid] += s_red[tid + s];
        __syncthreads();
    }
    if (tid == 0)
        conf_neg[b] = s_red[0] + (float)(k - s_cnt) * key2f(Tk);
}

// ---------------------------------------------------------------- kernel 3
// Single-wave f32 WMMA reduction of the 2304 partials: A = 64 values/iter,
// B = ones => D accumulates row-sums in full f32 precision (V_WMMA_F32_16X16X4_F32).
__global__ void final_reduce_wmma(const float* __restrict__ partials,
                                  float* __restrict__ out)
{
    const int lane = threadIdx.x;        // exactly 32 threads; EXEC all ones
    v8f acc = {};
    v2f ones; ones.x = 1.0f; ones.y = 1.0f;
    #pragma unroll 4
    for (int it = 0; it < NRED / 64; ++it) {
        float2 pv = ((const float2*)partials)[it * 32 + lane];
        v2f a; a.x = pv.x; a.y = pv.y;
        acc = __builtin_amdgcn_wmma_f32_16x16x4_f32(
                  false, a, false, ones, (short)0, acc, false, false);
    }
    // C layout: lane 0 holds (m=0..7, n=0), lane 16 holds (m=8..15, n=0)
    float s = acc[0] + acc[1] + acc[2] + acc[3] + acc[4] + acc[5] + acc[6] + acc[7];
    s += __shfl_xor(s, 16, 32);
    if (lane == 0) out[0] = s;
}

// ---------------------------------------------------------------- launcher
extern "C" void kernel_launch(void* const* d_in, const int* in_sizes, int n_in,
                              void* d_out, int out_size, void* d_ws, size_t ws_size,
                              hipStream_t stream) {
    const float* pred_loc = (const float*)d_in[0];
    const float* pred_cls = (const float*)d_in[1];
    const float* gt_boxes = (const float*)d_in[2];
    const int*   gt_labels= (const int*)  d_in[3];
    const float* dboxes   = (const float*)d_in[4];

    char* ws = (char*)d_ws;
    float* ranked   = (float*)ws;                                  // B*N floats
    int*   num_pos  = (int*)  (ws + (size_t)B_ * N_ * 4);          // 64 ints
    float* partials = (float*)(ws + (size_t)B_ * N_ * 4 + 256);    // NRED floats

    zero_numpos<<<1, 64, 0, stream>>>(num_pos);

    dim3 gA(NTILE, B_);
    ssd_match_kernel<<<gA, 256, 0, stream>>>(pred_loc, pred_cls, gt_boxes,
                                             gt_labels, dboxes,
                                             ranked, num_pos, partials);

    hard_neg_topk_kernel<<<B_, 1024, 0, stream>>>(ranked, num_pos,
                                                  partials + NPART);

    final_reduce_wmma<<<1, 32, 0, stream>>>(partials, (float*)d_out);
}